// ResidualBlock_21345987461187
// MI455X (gfx1250) — compile-verified
//
#include <hip/hip_runtime.h>
#include <hip/hip_bf16.h>
#include <math.h>

typedef float v2f __attribute__((ext_vector_type(2)));
typedef float v8f __attribute__((ext_vector_type(8)));

#define B_ 16
#define CIN 128
#define CMID 64
#define COUT 256
#define N_ 8192
#define S_ 2048
#define K_ 16
#define Q_ 16
#define BS (B_ * S_)
#define EPS 1e-5f

// ---------------------------------------------------------------------------
// Blocked fp32 WMMA GEMM:  Z[b, m, n] = sum_c W[m,c] * X[b, c, n] (+ bias[m])
// X layout (Btot, C, N) row-major.  Each wave computes a 64x16 output strip
// (4 row tiles sharing one B fragment per K-step -> 4x less X traffic).
// Block = 8 waves (256 threads) covering 8 adjacent column tiles.
// Uses V_WMMA_F32_16X16X4_F32 (wave32, 8-VGPR f32 accumulator).
// ---------------------------------------------------------------------------
__global__ __launch_bounds__(256)
void gemm_wmma_f32(const float* __restrict__ W, const float* __restrict__ X,
                   const float* __restrict__ bias, float* __restrict__ Z,
                   int M, int C, int N) {
  const int tc    = blockIdx.x * 8 + threadIdx.y;  // column tile (16 cols)
  const int strip = blockIdx.y;                    // 64-row strip
  const int col0 = tc * 16;
  const int b    = col0 / N;            // N multiple of 16 -> tile never crosses b
  const int n0   = col0 - b * N;
  const int m0   = strip * 64;
  const int lane = threadIdx.x;
  const int l15  = lane & 15;
  const int hi   = lane >> 4;           // 0: K pair {0,1}, 1: K pair {2,3}

  const float* Wr   = W + (size_t)(m0 + l15) * C;  // row tile 0; +16*C per tile
  const float* Xcol = X + (size_t)b * C * N + n0 + l15;

  v8f acc0 = {}, acc1 = {}, acc2 = {}, acc3 = {};
  for (int k0 = 0; k0 < C; k0 += 4) {
    const int ka = k0 + hi * 2;
    v2f bf;
    bf.x = Xcol[(size_t)ka * N];
    bf.y = Xcol[(size_t)(ka + 1) * N];
    if (k0 + 8 < C) {                   // prefetch X two K-steps ahead
      __builtin_prefetch(&Xcol[(size_t)(ka + 8) * N], 0, 1);
      __builtin_prefetch(&Xcol[(size_t)(ka + 9) * N], 0, 1);
    }
    v2f a;
    a.x = Wr[ka];              a.y = Wr[ka + 1];
    acc0 = __builtin_amdgcn_wmma_f32_16x16x4_f32(false, a, false, bf,
                                                 (short)0, acc0, false, false);
    a.x = Wr[16 * C + ka];     a.y = Wr[16 * C + ka + 1];
    acc1 = __builtin_amdgcn_wmma_f32_16x16x4_f32(false, a, false, bf,
                                                 (short)0, acc1, false, false);
    a.x = Wr[32 * C + ka];     a.y = Wr[32 * C + ka + 1];
    acc2 = __builtin_amdgcn_wmma_f32_16x16x4_f32(false, a, false, bf,
                                                 (short)0, acc2, false, false);
    a.x = Wr[48 * C + ka];     a.y = Wr[48 * C + ka + 1];
    acc3 = __builtin_amdgcn_wmma_f32_16x16x4_f32(false, a, false, bf,
                                                 (short)0, acc3, false, false);
  }

  const size_t outBase = (size_t)b * M * N + (size_t)n0 + l15;
  for (int v = 0; v < 8; ++v) {
    const int r0 = m0 + v + hi * 8;
    float v0 = acc0[v], v1 = acc1[v], v2 = acc2[v], v3 = acc3[v];
    if (bias) {
      v0 += bias[r0]; v1 += bias[r0 + 16]; v2 += bias[r0 + 32]; v3 += bias[r0 + 48];
    }
    Z[outBase + (size_t)r0 * N]        = v0;
    Z[outBase + (size_t)(r0 + 16) * N] = v1;
    Z[outBase + (size_t)(r0 + 32) * N] = v2;
    Z[outBase + (size_t)(r0 + 48) * N] = v3;
  }
}

// ---------------------------------------------------------------------------
// Per-channel train-mode norm statistics -> fused affine (scale, shift).
// element(ch, b, n) = Z[ch*chStride + b*bStride + n]
// scale = g * rsqrt(var + eps); shift = beta - mean*scale
// ---------------------------------------------------------------------------
__global__ __launch_bounds__(256)
void stats_kernel(const float* __restrict__ Z, int Bc, int Nn,
                  size_t chStride, size_t bStride,
                  const float* __restrict__ g, const float* __restrict__ bt,
                  int gmod, float* __restrict__ scale, float* __restrict__ shift) {
  const int ch = blockIdx.x;
  const float* base = Z + (size_t)ch * chStride;
  const long L = (long)Bc * Nn;
  float s = 0.f, s2 = 0.f;
  for (long i = threadIdx.x; i < L; i += blockDim.x) {
    const long bb = i / Nn, nn = i - bb * Nn;
    const float v = base[(size_t)bb * bStride + nn];
    s += v; s2 += v * v;
  }
  __shared__ float sh[256], sh2[256];
  sh[threadIdx.x] = s; sh2[threadIdx.x] = s2;
  __syncthreads();
  for (int o = 128; o > 0; o >>= 1) {
    if ((int)threadIdx.x < o) {
      sh[threadIdx.x]  += sh[threadIdx.x + o];
      sh2[threadIdx.x] += sh2[threadIdx.x + o];
    }
    __syncthreads();
  }
  if (threadIdx.x == 0) {
    const float mean = sh[0] / (float)L;
    const float var  = sh2[0] / (float)L - mean * mean;
    const float sc   = g[ch % gmod] * rsqrtf(var + EPS);
    scale[ch] = sc;
    shift[ch] = bt[ch % gmod] - mean * sc;
  }
}

// In-place affine (+ optional relu); ch = (i / chStride) % chMod
__global__ __launch_bounds__(256)
void apply_affine(float* __restrict__ Z, long total, long chStride, int chMod,
                  const float* __restrict__ scale, const float* __restrict__ shift,
                  int doRelu) {
  const long i = (long)blockIdx.x * blockDim.x + threadIdx.x;
  if (i >= total) return;
  const int ch = (int)((i / chStride) % chMod);
  float v = scale[ch] * Z[i] + shift[ch];
  if (doRelu) v = fmaxf(v, 0.f);
  Z[i] = v;
}

// ---------------------------------------------------------------------------
// Geometry + distance weights + fc1 raw.  One thread per (b, s).
//   pts = gather(pos) - sup ; d = |pts| ; dw = sigmoid(-a d + b), normalized
//   mat1raw[b,q,s,k] = fc1_w[q,:] . (pts / norm_radius)
// ---------------------------------------------------------------------------
__global__ __launch_bounds__(256)
void geom_fc1_kernel(const float* __restrict__ pos, const float* __restrict__ sup,
                     const int* __restrict__ idx, const float* __restrict__ fc1,
                     const float* __restrict__ alpha, const float* __restrict__ beta,
                     const float* __restrict__ nrad,
                     float* __restrict__ dwOut, float* __restrict__ mat1) {
  const int t = blockIdx.x * blockDim.x + threadIdx.x;
  if (t >= BS) return;
  const int b = t / S_, s = t - b * S_;
  const float a = alpha[0], be = beta[0], inr = 1.f / nrad[0];
  const float sx = sup[(size_t)b * 3 * S_ + 0 * S_ + s];
  const float sy = sup[(size_t)b * 3 * S_ + 1 * S_ + s];
  const float sz = sup[(size_t)b * 3 * S_ + 2 * S_ + s];
  const int* nid = idx + (size_t)t * K_;
  float dww[K_];
  float sum = 0.f;
  for (int k = 0; k < K_; ++k) {
    const int n = nid[k];
    const float px = pos[(size_t)b * 3 * N_ + 0 * N_ + n] - sx;
    const float py = pos[(size_t)b * 3 * N_ + 1 * N_ + n] - sy;
    const float pz = pos[(size_t)b * 3 * N_ + 2 * N_ + n] - sz;
    const float d = sqrtf(px * px + py * py + pz * pz);
    const float w = 1.f / (1.f + __expf(a * d - be));   // sigmoid(-a d + b)
    dww[k] = w; sum += w;
    const float qx = px * inr, qy = py * inr, qz = pz * inr;
    for (int q = 0; q < Q_; ++q) {
      const float r = fc1[q * 3 + 0] * qx + fc1[q * 3 + 1] * qy + fc1[q * 3 + 2] * qz;
      mat1[((size_t)(b * Q_ + q) * S_ + s) * K_ + k] = r;
    }
  }
  float ss = sum + ((sum == 0.f) ? 1.f : 0.f) + 1e-6f;
  const float inv = (float)K_ / ss;
  for (int k = 0; k < K_; ++k) dwOut[(size_t)t * K_ + k] = dww[k] * inv;
}

// ---------------------------------------------------------------------------
// Mid stage: matn1 = relu(IN1(mat1raw)); mp1 = max_k(matn1*dw);
// mat2raw = fc2 @ concat(matn1, mp1).  One thread per (b, s).
// ---------------------------------------------------------------------------
__global__ __launch_bounds__(256)
void mid_kernel(const float* __restrict__ mat1, const float* __restrict__ dw,
                const float* __restrict__ sc1, const float* __restrict__ sh1,
                const float* __restrict__ fc2, float* __restrict__ mat2) {
  const int t = blockIdx.x * blockDim.x + threadIdx.x;
  if (t >= BS) return;
  const int b = t / S_, s = t - b * S_;
  const float* dwv = dw + (size_t)t * K_;
  float mp[Q_];
  for (int q = 0; q < Q_; ++q) mp[q] = -3.4e38f;
  // pass 1: max-pool of normalized+relu'd mat1
  for (int k = 0; k < K_; ++k) {
    const float dk = dwv[k];
    for (int q = 0; q < Q_; ++q) {
      const int ch = b * Q_ + q;
      float v = sc1[ch] * mat1[((size_t)ch * S_ + s) * K_ + k] + sh1[ch];
      v = fmaxf(v, 0.f);
      const float vd = v * dk;
      if (vd > mp[q]) mp[q] = vd;
    }
  }
  // pass 2: fc2 over [matn1 ; mp1]
  for (int k = 0; k < K_; ++k) {
    float mn[Q_];
    for (int q = 0; q < Q_; ++q) {
      const int ch = b * Q_ + q;
      mn[q] = fmaxf(sc1[ch] * mat1[((size_t)ch * S_ + s) * K_ + k] + sh1[ch], 0.f);
    }
    for (int q2 = 0; q2 < Q_; ++q2) {
      float r = 0.f;
      for (int q = 0; q < Q_; ++q) r += fc2[q2 * 2 * Q_ + q] * mn[q];
      for (int q = 0; q < Q_; ++q) r += fc2[q2 * 2 * Q_ + Q_ + q] * mp[q];
      mat2[((size_t)(b * Q_ + q2) * S_ + s) * K_ + k] = r;
    }
  }
}

// ---------------------------------------------------------------------------
// Final mat stage + WMMA aggregation.  One wave per (b, s).
//   matn2 = relu(IN2(mat2raw)); mp2 = max_k(matn2*dw)
//   mat3  = relu(fc3 @ [matn2; mp2]) * dw          (16 x 16, staged in LDS)
//   agg(c,q) = sum_k feat(c,k) * mat3(q,k)         via 16 x WMMA f32 16x16x4
//   feat gathered on the fly from h0 (bn0+relu already applied).
//   AGG stored as (1024, B*S) row-major: row = c*16+q, col = b*S+s.
// ---------------------------------------------------------------------------
__global__ __launch_bounds__(32)
void mat3_agg_kernel(const float* __restrict__ mat2, const float* __restrict__ dw,
                     const float* __restrict__ sc2, const float* __restrict__ sh2,
                     const float* __restrict__ fc3, const float* __restrict__ h0,
                     const int* __restrict__ idx, float* __restrict__ AGG) {
  const int t = blockIdx.x;               // (b, s) site
  const int b = t / S_, s = t - b * S_;
  const int lane = threadIdx.x;
  const int l15 = lane & 15;
  const int hi  = lane >> 4;

  __shared__ float matbuf[Q_ * K_];       // matn2[q][k]
  __shared__ float mat3buf[Q_ * K_];      // mat3[q][k]
  __shared__ float mp2s[Q_];
  __shared__ float dws[K_];
  __shared__ int   nids[K_];

  if (lane < K_) {
    dws[lane]  = dw[(size_t)t * K_ + lane];
    nids[lane] = idx[(size_t)t * K_ + lane];
  }
  // matn2: 8 elements per lane
  for (int j = 0; j < 8; ++j) {
    const int e = lane * 8 + j;
    const int q = e >> 4, k = e & 15;
    const int ch = b * Q_ + q;
    matbuf[e] = fmaxf(sc2[ch] * mat2[((size_t)ch * S_ + s) * K_ + k] + sh2[ch], 0.f);
  }
  __syncthreads();
  if (lane < Q_) {
    float mx = -3.4e38f;
    for (int k = 0; k < K_; ++k) mx = fmaxf(mx, matbuf[lane * K_ + k] * dws[k]);
    mp2s[lane] = mx;
  }
  __syncthreads();
  for (int j = 0; j < 8; ++j) {
    const int e = lane * 8 + j;
    const int q = e >> 4, k = e & 15;
    float r = 0.f;
    for (int qp = 0; qp < Q_; ++qp) r += fc3[q * 2 * Q_ + qp] * matbuf[qp * K_ + k];
    for (int qp = 0; qp < Q_; ++qp) r += fc3[q * 2 * Q_ + Q_ + qp] * mp2s[qp];
    mat3buf[e] = fmaxf(r, 0.f) * dws[k];
  }
  __syncthreads();

  // WMMA: agg(c,q) = feat(c,k) x mat3^T(k,q), 4 c-tiles of 16, K=16
  const float* h0b = h0 + (size_t)b * CMID * N_;
  const size_t bs = (size_t)b * S_ + s;
  for (int ct = 0; ct < 4; ++ct) {
    const int c = ct * 16 + l15;
    v8f acc = {};
    for (int k0 = 0; k0 < K_; k0 += 4) {
      const int ka = k0 + hi * 2;
      v2f a, bf;
      a.x  = h0b[(size_t)c * N_ + nids[ka]];
      a.y  = h0b[(size_t)c * N_ + nids[ka + 1]];
      bf.x = mat3buf[l15 * K_ + ka];
      bf.y = mat3buf[l15 * K_ + ka + 1];
      acc = __builtin_amdgcn_wmma_f32_16x16x4_f32(false, a, false, bf,
                                                  (short)0, acc, false, false);
    }
    for (int v = 0; v < 8; ++v) {
      const int cr = ct * 16 + v + hi * 8;
      AGG[((size_t)cr * Q_ + l15) * (size_t)BS + bs] = acc[v];
    }
  }
}

// ---------------------------------------------------------------------------
// Final: out = relu( bn2(z2) + max_k bnsc(zsc[gathered]) )
// One thread per (b, o, s).
// ---------------------------------------------------------------------------
__global__ __launch_bounds__(256)
void final_kernel(const float* __restrict__ z2, const float* __restrict__ sc2bn,
                  const float* __restrict__ sh2bn, const float* __restrict__ zsc,
                  const float* __restrict__ scSC, const float* __restrict__ shSC,
                  const int* __restrict__ idx, float* __restrict__ out) {
  const long i = (long)blockIdx.x * blockDim.x + threadIdx.x;
  if (i >= (long)B_ * COUT * S_) return;
  const int b = (int)(i / ((long)COUT * S_));
  const int r = (int)(i - (long)b * COUT * S_);
  const int o = r / S_, s = r - o * S_;
  const float h = sc2bn[o] * z2[(size_t)o * BS + (size_t)b * S_ + s] + sh2bn[o];
  const float a = scSC[o], c = shSC[o];
  const int* nid = idx + ((size_t)b * S_ + s) * K_;
  const float* zb = zsc + ((size_t)b * COUT + o) * N_;
  float mx = -3.4e38f;
  for (int k = 0; k < K_; ++k) mx = fmaxf(mx, a * zb[nid[k]] + c);
  out[((size_t)b * COUT + o) * S_ + s] = fmaxf(h + mx, 0.f);
}

// ---------------------------------------------------------------------------
extern "C" void kernel_launch(void* const* d_in, const int* in_sizes, int n_in,
                              void* d_out, int out_size, void* d_ws, size_t ws_size,
                              hipStream_t stream) {
  const float* x     = (const float*)d_in[0];
  const float* pos   = (const float*)d_in[1];
  const float* sup   = (const float*)d_in[2];
  const int*   idx   = (const int*)d_in[3];
  // d_in[4] = mask_indices (unused, as in reference)
  const float* cv0_w = (const float*)d_in[5];
  const float* cv0_b = (const float*)d_in[6];
  const float* bn0_g = (const float*)d_in[7];
  const float* bn0_b = (const float*)d_in[8];
  const float* fc1_w = (const float*)d_in[9];
  const float* fc2_w = (const float*)d_in[10];
  const float* fc3_w = (const float*)d_in[11];
  const float* in1_g = (const float*)d_in[12];
  const float* in1_b = (const float*)d_in[13];
  const float* in2_g = (const float*)d_in[14];
  const float* in2_b = (const float*)d_in[15];
  const float* alpha = (const float*)d_in[16];
  const float* beta  = (const float*)d_in[17];
  const float* nrad  = (const float*)d_in[18];
  const float* cv_w  = (const float*)d_in[19];
  const float* bn1_g = (const float*)d_in[20];
  const float* bn1_b = (const float*)d_in[21];
  const float* cv2_w = (const float*)d_in[22];
  const float* cv2_b = (const float*)d_in[23];
  const float* bn2_g = (const float*)d_in[24];
  const float* bn2_b = (const float*)d_in[25];
  const float* sc_w  = (const float*)d_in[26];
  const float* sc_b  = (const float*)d_in[27];
  const float* bnsc_g = (const float*)d_in[28];
  const float* bnsc_b = (const float*)d_in[29];
  float* out = (float*)d_out;

  // workspace carve (floats)
  float* p = (float*)d_ws;
  float* z0   = p; p += (size_t)B_ * CMID * N_;   // 8.4M  (conv0 raw -> h0 in place)
  float* zsc  = p; p += (size_t)B_ * COUT * N_;   // 33.6M (shortcut conv raw)
  float* dwb  = p; p += (size_t)BS * K_;          // 0.5M
  float* mat1 = p; p += (size_t)B_ * Q_ * S_ * K_;// 8.4M
  float* mat2 = p; p += (size_t)B_ * Q_ * S_ * K_;// 8.4M
  float* AGG  = p; p += (size_t)CMID * Q_ * BS;   // 33.6M (1024 x 32768)
  float* zf   = p; p += (size_t)CMID * BS;        // 2.1M  (fka out -> h1 in place)
  float* z2   = p; p += (size_t)COUT * BS;        // 8.4M
  float* sc0  = p; p += CMID;   float* sh0  = p; p += CMID;
  float* scS  = p; p += COUT;   float* shS  = p; p += COUT;
  float* sc1  = p; p += B_*Q_;  float* sh1  = p; p += B_*Q_;
  float* sc2i = p; p += B_*Q_;  float* sh2i = p; p += B_*Q_;
  float* scB1 = p; p += CMID;   float* shB1 = p; p += CMID;
  float* scB2 = p; p += COUT;   float* shB2 = p; p += COUT;

  const dim3 blk(32, 8);

  // 1) cv0: z0(B,64,N) = cv0_w(64,128) @ x(B,128,N) + b   (64-row strip: 1)
  gemm_wmma_f32<<<dim3((B_ * N_) / 16 / 8, CMID / 64), blk, 0, stream>>>(
      cv0_w, x, cv0_b, z0, CMID, CIN, N_);
  // 2) bn0 stats -> affine
  stats_kernel<<<CMID, 256, 0, stream>>>(z0, B_, N_, N_, (size_t)CMID * N_,
                                         bn0_g, bn0_b, CMID, sc0, sh0);
  // 3) z0 <- h0 = relu(bn0(z0))
  {
    long tot = (long)B_ * CMID * N_;
    apply_affine<<<(tot + 255) / 256, 256, 0, stream>>>(z0, tot, N_, CMID, sc0, sh0, 1);
  }
  // 4) shortcut conv: zsc(B,256,N) = sc_w(256,128) @ x + b
  gemm_wmma_f32<<<dim3((B_ * N_) / 16 / 8, COUT / 64), blk, 0, stream>>>(
      sc_w, x, sc_b, zsc, COUT, CIN, N_);
  // 5) bnsc stats (affine fused into final gather-max)
  stats_kernel<<<COUT, 256, 0, stream>>>(zsc, B_, N_, N_, (size_t)COUT * N_,
                                         bnsc_g, bnsc_b, COUT, scS, shS);
  // 6) geometry + dw + fc1 raw
  geom_fc1_kernel<<<(BS + 255) / 256, 256, 0, stream>>>(pos, sup, idx, fc1_w,
                                                        alpha, beta, nrad, dwb, mat1);
  // 7) IN1 stats (channel = b*Q+q over S*K)
  stats_kernel<<<B_ * Q_, 256, 0, stream>>>(mat1, 1, S_ * K_, (size_t)S_ * K_, 0,
                                            in1_g, in1_b, Q_, sc1, sh1);
  // 8) mid stage (fc2)
  mid_kernel<<<(BS + 255) / 256, 256, 0, stream>>>(mat1, dwb, sc1, sh1, fc2_w, mat2);
  // 9) IN2 stats
  stats_kernel<<<B_ * Q_, 256, 0, stream>>>(mat2, 1, S_ * K_, (size_t)S_ * K_, 0,
                                            in2_g, in2_b, Q_, sc2i, sh2i);
  // 10) mat3 + WMMA aggregation -> AGG(1024, B*S)
  mat3_agg_kernel<<<BS, 32, 0, stream>>>(mat2, dwb, sc2i, sh2i, fc3_w, z0, idx, AGG);
  // 11) cv_w contraction: zf(64, B*S) = cv_w(64,1024) @ AGG(1024, B*S)
  gemm_wmma_f32<<<dim3(BS / 16 / 8, CMID / 64), blk, 0, stream>>>(
      cv_w, AGG, nullptr, zf, CMID, CMID * Q_, BS);
  // 12) bn1 stats + apply relu in place
  stats_kernel<<<CMID, 256, 0, stream>>>(zf, 1, BS, BS, 0, bn1_g, bn1_b, CMID, scB1, shB1);
  {
    long tot = (long)CMID * BS;
    apply_affine<<<(tot + 255) / 256, 256, 0, stream>>>(zf, tot, BS, CMID, scB1, shB1, 1);
  }
  // 13) cv2: z2(256, B*S) = cv2_w(256,64) @ zf(64, B*S) + b
  gemm_wmma_f32<<<dim3(BS / 16 / 8, COUT / 64), blk, 0, stream>>>(
      cv2_w, zf, cv2_b, z2, COUT, CMID, BS);
  // 14) bn2 stats
  stats_kernel<<<COUT, 256, 0, stream>>>(z2, 1, BS, BS, 0, bn2_g, bn2_b, COUT, scB2, shB2);
  // 15) residual: out = relu(bn2(z2) + gather_max(bnsc(zsc)))
  {
    long tot = (long)B_ * COUT * S_;
    final_kernel<<<(tot + 255) / 256, 256, 0, stream>>>(z2, scB2, shB2, zsc, scS, shS,
                                                        idx, out);
  }
}